// MtpModuleANE_28784870818406
// MI455X (gfx1250) — compile-verified
//
#include <hip/hip_runtime.h>
#include <hip/hip_bf16.h>
#include <math.h>

// ---------------- model constants ----------------
#define H_DIM   2304
#define I_FF    9216
#define NH      8
#define NKV     4
#define HD      256
#define W_CTX   4096
#define V_VOC   65536
#define EPS_F   1e-6f
#define SOFTCAP 30.0f
#define HALF_HD 128

typedef __attribute__((ext_vector_type(2))) float v2f;
typedef __attribute__((ext_vector_type(8))) float v8f;

// =====================================================================
// GEMV via V_WMMA_F32_16X16X4_F32.
// y[r] = sum_k W[r,k]*x[k]  (W row-major, R x K). Wave = 16 rows.
// A frag: lanes 0-15 carry rows (slots 0,1 in a.x/a.y), lanes 16-31 the
// same rows at slots 2,3. Slot->K mapping is arbitrary if A and B agree,
// so per 16-K step: hi=0 lanes load float4 at k, hi=1 lanes at k+8:
//   wmma0 slots {k,k+1 | k+8,k+9}   from f4.xy
//   wmma1 slots {k+2,k+3 | k+10,k+11} from f4.zw
//   second float4 at k+4 (hi=0) / k+12 (hi=1) feeds wmma2/wmma3.
// -> 4x global_load_b128 (2 W, 2 x) + 4x wmma per 16 K per lane.
// D frag: VGPR r -> lane0 has (M=r,N=0), lane16 has (M=8+r,N=0).
// grid.x = R/128 (8 waves * 16 rows), grid.y = split-K index.
// Requires: kLen % 16 == 0, K % 4 == 0 (alignment), R % 128 == 0.
// =====================================================================
__global__ __launch_bounds__(256) void gemv_wmma_f32(
    const float* __restrict__ W, const float* __restrict__ x,
    float* __restrict__ partial, int R, int K, int kLen)
{
    const int tid  = threadIdx.x;
    const int wave = tid >> 5;
    const int lane = tid & 31;
    const int rowBase = blockIdx.x * 128 + wave * 16;
    const int row  = rowBase + (lane & 15);
    const int hi   = lane >> 4;                 // 0 or 1
    const int k0   = blockIdx.y * kLen;
    int kEnd = k0 + kLen; if (kEnd > K) kEnd = K;

    const float* wrow = W + (size_t)row * K;
    const int sl = hi << 3;                     // +8 elements for hi lanes

#if __has_builtin(__builtin_amdgcn_wmma_f32_16x16x4_f32)
    v8f c = {0.f,0.f,0.f,0.f,0.f,0.f,0.f,0.f};
    #pragma unroll 2
    for (int k = k0; k < kEnd; k += 16) {
        const float4 wa = *(const float4*)(wrow + k + sl);
        const float4 wb = *(const float4*)(wrow + k + sl + 4);
        const float4 xa = *(const float4*)(x    + k + sl);
        const float4 xb = *(const float4*)(x    + k + sl + 4);
        v2f a, b;
        a.x = wa.x; a.y = wa.y; b.x = xa.x; b.y = xa.y;
        c = __builtin_amdgcn_wmma_f32_16x16x4_f32(false, a, false, b, (short)0, c, false, false);
        a.x = wa.z; a.y = wa.w; b.x = xa.z; b.y = xa.w;
        c = __builtin_amdgcn_wmma_f32_16x16x4_f32(false, a, false, b, (short)0, c, false, false);
        a.x = wb.x; a.y = wb.y; b.x = xb.x; b.y = xb.y;
        c = __builtin_amdgcn_wmma_f32_16x16x4_f32(false, a, false, b, (short)0, c, false, false);
        a.x = wb.z; a.y = wb.w; b.x = xb.z; b.y = xb.w;
        c = __builtin_amdgcn_wmma_f32_16x16x4_f32(false, a, false, b, (short)0, c, false, false);
    }
    float* out = partial + (size_t)blockIdx.y * R + rowBase;
    if (lane == 0) {
        #pragma unroll
        for (int m = 0; m < 8; ++m) out[m] = c[m];
    } else if (lane == 16) {
        #pragma unroll
        for (int m = 0; m < 8; ++m) out[8 + m] = c[m];
    }
#else
    // VALU fallback: lane m and lane m+16 each hold half the K stream of row m.
    float s = 0.f;
    #pragma unroll 2
    for (int k = k0; k < kEnd; k += 16) {
        const float4 wa = *(const float4*)(wrow + k + sl);
        const float4 wb = *(const float4*)(wrow + k + sl + 4);
        const float4 xa = *(const float4*)(x    + k + sl);
        const float4 xb = *(const float4*)(x    + k + sl + 4);
        s = fmaf(wa.x, xa.x, s); s = fmaf(wa.y, xa.y, s);
        s = fmaf(wa.z, xa.z, s); s = fmaf(wa.w, xa.w, s);
        s = fmaf(wb.x, xb.x, s); s = fmaf(wb.y, xb.y, s);
        s = fmaf(wb.z, xb.z, s); s = fmaf(wb.w, xb.w, s);
    }
    float tot = s + __shfl_xor(s, 16, 32);
    if (lane < 16) partial[(size_t)blockIdx.y * R + rowBase + (lane & 15)] = tot;
#endif
}

// fixed-order split-K reduction: y[i] = sum_j partial[j*R + i]
__global__ void reduce_partials_k(const float* __restrict__ p,
                                  float* __restrict__ y, int R, int S)
{
    int i = blockIdx.x * 256 + threadIdx.x;
    if (i < R) {
        float s = 0.f;
        for (int j = 0; j < S; ++j) s += p[(size_t)j * R + i];
        y[i] = s;
    }
}

// rmsnorm over n elems (single block); optional residual add; optional
// secondary destination (for writing hidden_out both aligned and to d_out).
__global__ __launch_bounds__(256) void rmsnorm_k(
    const float* __restrict__ x, const float* __restrict__ w,
    const float* __restrict__ resid, float* __restrict__ y,
    float* __restrict__ y2, int n)
{
    __shared__ float red[256];
    int t = threadIdx.x;
    float s = 0.f;
    for (int i = t; i < n; i += 256) { float v = x[i]; s += v * v; }
    red[t] = s; __syncthreads();
    for (int off = 128; off > 0; off >>= 1) {
        if (t < off) red[t] += red[t + off];
        __syncthreads();
    }
    float inv = rsqrtf(red[0] / (float)n + EPS_F);
    for (int i = t; i < n; i += 256) {
        float v = x[i] * inv * w[i];
        if (resid) v += resid[i];
        y[i] = v;
        if (y2) y2[i] = v;
    }
}

// per-head rmsnorm + RoPE for q (blocks 0..7) and k (blocks 8..11).
__global__ __launch_bounds__(256) void qknorm_rope_k(
    const float* __restrict__ q, const float* __restrict__ k,
    const float* __restrict__ qw, const float* __restrict__ kw,
    const float* __restrict__ cosv, const float* __restrict__ sinv,
    float* __restrict__ q_rot, float* __restrict__ k_rot)
{
    __shared__ float red[256];
    __shared__ float buf[256];
    int t = threadIdx.x;
    int b = blockIdx.x;
    const float* src; const float* nw; float* dst;
    if (b < NH) { src = q + b * HD;        nw = qw; dst = q_rot + b * HD; }
    else        { src = k + (b - NH) * HD; nw = kw; dst = k_rot + (b - NH) * HD; }
    float v = src[t];
    red[t] = v * v; __syncthreads();
    for (int off = 128; off > 0; off >>= 1) {
        if (t < off) red[t] += red[t + off];
        __syncthreads();
    }
    float inv = rsqrtf(red[0] / (float)HD + EPS_F);
    buf[t] = v * inv * nw[t];
    __syncthreads();
    if (t < HALF_HD) {
        float c = cosv[t], s = sinv[t];
        float x1 = buf[t], x2 = buf[t + HALF_HD];
        dst[t]           = x1 * c - x2 * s;
        dst[t + HALF_HD] = x2 * c + x1 * s;
    }
}

// kv_out = kv_in*(1-u) + new*u on rows where update_idx != 0
// (bulk copy done via memcpyAsync beforehand). grid = W_CTX, threads = HD.
__global__ void cache_blend_k(
    const float* __restrict__ kin, const float* __restrict__ vin,
    const float* __restrict__ upd, const float* __restrict__ k_rot,
    const float* __restrict__ v_vec,
    float* __restrict__ kout, float* __restrict__ vout)
{
    int w = blockIdx.x;
    float u = upd[w];
    if (u == 0.f) return;
    float keep = 1.f - u;
    int t = threadIdx.x;
    for (int n = 0; n < NKV; ++n) {
        size_t idx = ((size_t)n * W_CTX + w) * HD + t;
        kout[idx] = kin[idx] * keep + k_rot[n * HD + t] * u;
        vout[idx] = vin[idx] * keep + v_vec[n * HD + t] * u;
    }
}

// attention, one block per q head. scores in LDS, softmax, PV coalesced.
__global__ __launch_bounds__(256) void attn_k(
    const float* __restrict__ qr, const float* __restrict__ Kc,
    const float* __restrict__ Vc, const float* __restrict__ mask,
    float* __restrict__ out)
{
    __shared__ float sc[W_CTX];
    __shared__ float red[256];
    __shared__ __align__(16) float qs[HD];
    int t = threadIdx.x;
    int h = blockIdx.x;
    int kvh = h >> 1;                       // NH/NKV == 2
    qs[t] = qr[h * HD + t];
    __syncthreads();

    const float* Kbase = Kc + (size_t)kvh * W_CTX * HD;
    float lmax = -3.0e38f;
    for (int k = t; k < W_CTX; k += 256) {
        const float4* kr = (const float4*)(Kbase + (size_t)k * HD);
        const float4* q4 = (const float4*)qs;
        float d = 0.f;
        #pragma unroll 8
        for (int j = 0; j < HD / 4; ++j) {
            float4 kv4 = kr[j], qv4 = q4[j];
            d += kv4.x * qv4.x + kv4.y * qv4.y + kv4.z * qv4.z + kv4.w * qv4.w;
        }
        float s = d * 0.0625f + mask[k];    // 1/sqrt(256)
        sc[k] = s;
        lmax = fmaxf(lmax, s);
    }
    red[t] = lmax; __syncthreads();
    for (int off = 128; off > 0; off >>= 1) {
        if (t < off) red[t] = fmaxf(red[t], red[t + off]);
        __syncthreads();
    }
    float m = red[0];
    __syncthreads();
    float lsum = 0.f;
    for (int k = t; k < W_CTX; k += 256) {
        float p = expf(sc[k] - m);
        sc[k] = p;
        lsum += p;
    }
    red[t] = lsum; __syncthreads();
    for (int off = 128; off > 0; off >>= 1) {
        if (t < off) red[t] += red[t + off];
        __syncthreads();
    }
    float invs = 1.f / red[0];

    const float* Vbase = Vc + (size_t)kvh * W_CTX * HD;
    float acc = 0.f;
    #pragma unroll 8
    for (int k = 0; k < W_CTX; ++k)
        acc = fmaf(sc[k], Vbase[(size_t)k * HD + t], acc);
    out[h * HD + t] = acc * invs;
}

// exact GELU(gate) * up
__global__ void gelu_mul_k(const float* __restrict__ g,
                           const float* __restrict__ u,
                           float* __restrict__ o, int n)
{
    int i = blockIdx.x * 256 + threadIdx.x;
    if (i < n) {
        float x = g[i];
        float ge = 0.5f * x * (1.f + erff(x * 0.70710678118654752f));
        o[i] = ge * u[i];
    }
}

// argmax stage 1: each block reduces 256 contiguous logits (lowest-index ties)
__global__ __launch_bounds__(256) void argmax1_k(
    const float* __restrict__ logits, float* __restrict__ bv, int* __restrict__ bi)
{
    __shared__ float rv[256];
    __shared__ int   ri[256];
    int t = threadIdx.x;
    int g = blockIdx.x * 256 + t;
    rv[t] = logits[g]; ri[t] = g; __syncthreads();
    for (int off = 128; off > 0; off >>= 1) {
        if (t < off) {
            float v2 = rv[t + off]; int i2 = ri[t + off];
            if (v2 > rv[t] || (v2 == rv[t] && i2 < ri[t])) { rv[t] = v2; ri[t] = i2; }
        }
        __syncthreads();
    }
    if (t == 0) { bv[blockIdx.x] = rv[0]; bi[blockIdx.x] = ri[0]; }
}

// argmax stage 2 over 256 block results; apply softcap to the winner only.
__global__ __launch_bounds__(256) void argmax2_k(
    const float* __restrict__ bv, const int* __restrict__ bi, float* __restrict__ dout)
{
    __shared__ float rv[256];
    __shared__ int   ri[256];
    int t = threadIdx.x;
    rv[t] = bv[t]; ri[t] = bi[t]; __syncthreads();
    for (int off = 128; off > 0; off >>= 1) {
        if (t < off) {
            float v2 = rv[t + off]; int i2 = ri[t + off];
            if (v2 > rv[t] || (v2 == rv[t] && i2 < ri[t])) { rv[t] = v2; ri[t] = i2; }
        }
        __syncthreads();
    }
    if (t == 0) {
        ((int*)dout)[0] = ri[0];
        dout[1] = tanhf(rv[0] / SOFTCAP) * SOFTCAP;
    }
}

// ---------------- workspace layout (float offsets, all 16B aligned) --------
#define WS_P    0         // split-K partials, max 16*2304 = 36864
#define WS_X    40960     // 2304
#define WS_HN   43264     // 2304
#define WS_QV   45568     // 2048
#define WS_KV   47616     // 1024
#define WS_VV   48640     // 1024
#define WS_QR   49664     // 2048
#define WS_KR   51712     // 1024
#define WS_AO   52736     // 2048
#define WS_OO   54784     // 2304
#define WS_X2   57088     // 2304
#define WS_HF   59392     // 2304
#define WS_GA   61696     // 9216
#define WS_UP   70912     // 9216
#define WS_DA   80128     // 9216
#define WS_DN   89344     // 2304
#define WS_X3   91648     // 2304
#define WS_LG   93952     // 65536
#define WS_BV   159488    // 256
#define WS_BI   159744    // 256 ints
#define WS_HO   160000    // 2304 (aligned copy of hidden_out for lm_head GEMV)

// d_out layout (floats): [0]=index bits, [1]=top value, [2..]=hidden_out(2304),
// then kv_k_out (4*4096*256), then kv_v_out.
#define OUT_HID 2
#define OUT_KO  (2 + H_DIM)
#define OUT_VO  (OUT_KO + NKV * W_CTX * HD)

extern "C" void kernel_launch(void* const* d_in, const int* in_sizes, int n_in,
                              void* d_out, int out_size, void* d_ws, size_t ws_size,
                              hipStream_t stream)
{
    const float* hidden_prev   = (const float*)d_in[0];
    const float* embed_token   = (const float*)d_in[1];
    const float* kv_k_in       = (const float*)d_in[2];
    const float* kv_v_in       = (const float*)d_in[3];
    const float* cosv          = (const float*)d_in[4];
    const float* sinv          = (const float*)d_in[5];
    const float* mask          = (const float*)d_in[6];
    const float* update_idx    = (const float*)d_in[7];
    const float* in_h_w        = (const float*)d_in[8];
    const float* in_e_w        = (const float*)d_in[9];
    const float* input_ln_w    = (const float*)d_in[10];
    const float* post_attn_ln  = (const float*)d_in[11];
    const float* pre_ffn_ln    = (const float*)d_in[12];
    const float* post_ffn_ln   = (const float*)d_in[13];
    const float* final_ln      = (const float*)d_in[14];
    const float* q_w           = (const float*)d_in[15];
    const float* k_w           = (const float*)d_in[16];
    const float* v_w           = (const float*)d_in[17];
    const float* o_w           = (const float*)d_in[18];
    const float* q_norm_w      = (const float*)d_in[19];
    const float* k_norm_w      = (const float*)d_in[20];
    const float* gate_w        = (const float*)d_in[21];
    const float* up_w          = (const float*)d_in[22];
    const float* down_w        = (const float*)d_in[23];
    const float* lm_head_w     = (const float*)d_in[24];

    float* ws   = (float*)d_ws;
    float* dout = (float*)d_out;
    float* P    = ws + WS_P;

    // 1) x = in_h_w @ h + in_e_w @ e  (split-K 4 each, 8 partial slots)
    gemv_wmma_f32<<<dim3(18, 4), 256, 0, stream>>>(in_h_w, hidden_prev, P,              H_DIM, H_DIM, 576);
    gemv_wmma_f32<<<dim3(18, 4), 256, 0, stream>>>(in_e_w, embed_token, P + 4 * H_DIM,  H_DIM, H_DIM, 576);
    reduce_partials_k<<<9, 256, 0, stream>>>(P, ws + WS_X, H_DIM, 8);

    // 2) hn = rmsnorm(x, input_ln)
    rmsnorm_k<<<1, 256, 0, stream>>>(ws + WS_X, input_ln_w, nullptr, ws + WS_HN, nullptr, H_DIM);

    // 3) q/k/v projections
    gemv_wmma_f32<<<dim3(16, 4), 256, 0, stream>>>(q_w, ws + WS_HN, P,         2048, H_DIM, 576);
    gemv_wmma_f32<<<dim3( 8, 4), 256, 0, stream>>>(k_w, ws + WS_HN, P + 8192,  1024, H_DIM, 576);
    gemv_wmma_f32<<<dim3( 8, 4), 256, 0, stream>>>(v_w, ws + WS_HN, P + 12288, 1024, H_DIM, 576);
    reduce_partials_k<<<8, 256, 0, stream>>>(P,         ws + WS_QV, 2048, 4);
    reduce_partials_k<<<4, 256, 0, stream>>>(P + 8192,  ws + WS_KV, 1024, 4);
    reduce_partials_k<<<4, 256, 0, stream>>>(P + 12288, ws + WS_VV, 1024, 4);

    // 4) per-head q/k rmsnorm + rope
    qknorm_rope_k<<<12, 256, 0, stream>>>(ws + WS_QV, ws + WS_KV, q_norm_w, k_norm_w,
                                          cosv, sinv, ws + WS_QR, ws + WS_KR);

    // 5) kv cache: bulk copy then blend updated row(s)
    size_t kvBytes = (size_t)NKV * W_CTX * HD * sizeof(float);
    hipMemcpyAsync(dout + OUT_KO, kv_k_in, kvBytes, hipMemcpyDeviceToDevice, stream);
    hipMemcpyAsync(dout + OUT_VO, kv_v_in, kvBytes, hipMemcpyDeviceToDevice, stream);
    cache_blend_k<<<W_CTX, HD, 0, stream>>>(kv_k_in, kv_v_in, update_idx,
                                            ws + WS_KR, ws + WS_VV,
                                            dout + OUT_KO, dout + OUT_VO);

    // 6) attention (reads updated caches from d_out)
    attn_k<<<NH, 256, 0, stream>>>(ws + WS_QR, dout + OUT_KO, dout + OUT_VO, mask, ws + WS_AO);

    // 7) o-proj, post-attn norm + residual
    gemv_wmma_f32<<<dim3(18, 4), 256, 0, stream>>>(o_w, ws + WS_AO, P, H_DIM, 2048, 512);
    reduce_partials_k<<<9, 256, 0, stream>>>(P, ws + WS_OO, H_DIM, 4);
    rmsnorm_k<<<1, 256, 0, stream>>>(ws + WS_OO, post_attn_ln, ws + WS_X, ws + WS_X2, nullptr, H_DIM);

    // 8) FFN
    rmsnorm_k<<<1, 256, 0, stream>>>(ws + WS_X2, pre_ffn_ln, nullptr, ws + WS_HF, nullptr, H_DIM);
    gemv_wmma_f32<<<dim3(72, 2), 256, 0, stream>>>(gate_w, ws + WS_HF, P,             I_FF, H_DIM, 1152);
    gemv_wmma_f32<<<dim3(72, 2), 256, 0, stream>>>(up_w,   ws + WS_HF, P + 2 * I_FF,  I_FF, H_DIM, 1152);
    reduce_partials_k<<<36, 256, 0, stream>>>(P,            ws + WS_GA, I_FF, 2);
    reduce_partials_k<<<36, 256, 0, stream>>>(P + 2 * I_FF, ws + WS_UP, I_FF, 2);
    gelu_mul_k<<<36, 256, 0, stream>>>(ws + WS_GA, ws + WS_UP, ws + WS_DA, I_FF);
    gemv_wmma_f32<<<dim3(18, 16), 256, 0, stream>>>(down_w, ws + WS_DA, P, H_DIM, I_FF, 576);
    reduce_partials_k<<<9, 256, 0, stream>>>(P, ws + WS_DN, H_DIM, 16);
    rmsnorm_k<<<1, 256, 0, stream>>>(ws + WS_DN, post_ffn_ln, ws + WS_X2, ws + WS_X3, nullptr, H_DIM);

    // 9) hidden_out = rmsnorm(x3, final_ln) -> aligned ws copy + d_out slice
    rmsnorm_k<<<1, 256, 0, stream>>>(ws + WS_X3, final_ln, nullptr, ws + WS_HO, dout + OUT_HID, H_DIM);

    // 10) logits from the aligned copy (raw; softcap is monotonic)
    gemv_wmma_f32<<<dim3(512, 1), 256, 0, stream>>>(lm_head_w, ws + WS_HO,
                                                    ws + WS_LG, V_VOC, H_DIM, H_DIM);

    // 11) argmax + softcapped top value
    argmax1_k<<<256, 256, 0, stream>>>(ws + WS_LG, ws + WS_BV, (int*)(ws + WS_BI));
    argmax2_k<<<1, 256, 0, stream>>>(ws + WS_BV, (const int*)(ws + WS_BI), dout);
}